// EGNN_20744692040307
// MI455X (gfx1250) — compile-verified
//
#include <hip/hip_runtime.h>
#include <hip/hip_bf16.h>
#include <math.h>

// ---------------------------------------------------------------------------
// Problem constants (from reference)
// ---------------------------------------------------------------------------
#define NN      8192
#define IN_DIM  128
#define HID     256
#define SLOPE   0.01f
#define HEADROWS 32            // NN / HID = rows of Ep actually needed for v

typedef __attribute__((ext_vector_type(2))) float v2f;
typedef __attribute__((ext_vector_type(4))) float f4;
typedef __attribute__((ext_vector_type(8))) float v8f;

#if defined(__has_builtin)
#  if __has_builtin(__builtin_amdgcn_wmma_f32_16x16x4_f32)
#    define HAVE_WMMA_F32 1
#  endif
#endif
#ifndef HAVE_WMMA_F32
#  define HAVE_WMMA_F32 0
#endif

// D(16x16) = A(16x4) * B(4x16) + C, fp32, wave32.
// Lane layout (ISA 7.12.2): A: vgpr j holds K=(lane/16)*2+j for M=lane%16.
// B (symmetric guess): vgpr j holds K=(lane/16)*2+j for N=lane%16.
// C/D: vgpr r holds (M = r + (lane/16)*8, N = lane%16).
__device__ __forceinline__ v8f wmma4(v2f a, v2f b, v8f c) {
#if HAVE_WMMA_F32
  return __builtin_amdgcn_wmma_f32_16x16x4_f32(
      /*neg_a=*/false, a, /*neg_b=*/false, b,
      /*c_mod=*/(short)0, c, /*reuse_a=*/false, /*reuse_b=*/false);
#else
  // Cross-lane emulation of the same layout (keeps compile clean if the
  // builtin is unavailable; histogram reveals which path was taken).
  const int lane = threadIdx.x & 31;
  const int n    = lane & 15;
  v8f d;
#pragma unroll
  for (int r = 0; r < 8; ++r) {
    const int M = r + (lane >> 4) * 8;
    float acc = c[r];
#pragma unroll
    for (int k = 0; k < 4; ++k) {
      float av = __shfl((k & 1) ? a[1] : a[0], (k >> 1) * 16 + M, 32);
      float bv = __shfl((k & 1) ? b[1] : b[0], (k >> 1) * 16 + n, 32);
      acc = fmaf(av, bv, acc);
    }
    d[r] = acc;
  }
  return d;
#endif
}

// ---------------------------------------------------------------------------
// K0: s = sum(Ai) + sum(Aj)        (256 elements total)
// ---------------------------------------------------------------------------
__global__ void k_scalar(const float* __restrict__ Ai,
                         const float* __restrict__ Aj,
                         float* __restrict__ sbuf) {
  __shared__ float red[256];
  const int tid = threadIdx.x;
  red[tid] = (tid < 128) ? Ai[tid] : Aj[tid - 128];
  __syncthreads();
  for (int o = 128; o > 0; o >>= 1) {
    if (tid < o) red[tid] += red[tid + o];
    __syncthreads();
  }
  if (tid == 0) sbuf[0] = red[0];
}

// ---------------------------------------------------------------------------
// K1: g = exp(leaky_relu(s * (X @ W^T + b)))   via fp32 WMMA, one wave/tile
//     grid (NN/16, HID/16), block 32
// ---------------------------------------------------------------------------
__global__ void k_gemm1(const float* __restrict__ X,
                        const float* __restrict__ W,
                        const float* __restrict__ b,
                        const float* __restrict__ sbuf,
                        float* __restrict__ g) {
  const int m0   = blockIdx.x * 16;
  const int n0   = blockIdx.y * 16;
  const int lane = threadIdx.x;
  const int half = lane >> 4;   // 0 or 1
  const int l16  = lane & 15;
  const int n    = n0 + l16;
  const float s  = sbuf[0];
  const float bn = b[n];

  v8f acc;
#pragma unroll
  for (int r = 0; r < 8; ++r) acc[r] = bn;   // C pre-seeded with bias

  const float* __restrict__ xr = X + (size_t)(m0 + l16) * IN_DIM;
  const float* __restrict__ wr = W + (size_t)n * IN_DIM;
#pragma unroll 4
  for (int k0 = 0; k0 < IN_DIM; k0 += 4) {
    v2f a, bb;
    a[0]  = xr[k0 + half * 2 + 0];
    a[1]  = xr[k0 + half * 2 + 1];
    bb[0] = wr[k0 + half * 2 + 0];
    bb[1] = wr[k0 + half * 2 + 1];
    acc = wmma4(a, bb, acc);
  }
#pragma unroll
  for (int r = 0; r < 8; ++r) {
    const int m = m0 + r + half * 8;
    float v  = acc[r] * s;
    float lr = v > 0.0f ? v : SLOPE * v;
    g[(size_t)m * HID + n] = expf(lr);
  }
}

// ---------------------------------------------------------------------------
// K2: Ep_head = E[0:32,:] @ g      (32 x 256, K = 8192), fp32 WMMA
//     grid (2, HID/16), block 32
// ---------------------------------------------------------------------------
__global__ void k_gemm2(const float* __restrict__ E,
                        const float* __restrict__ g,
                        float* __restrict__ eph) {
  const int m0   = blockIdx.x * 16;
  const int n0   = blockIdx.y * 16;
  const int lane = threadIdx.x;
  const int half = lane >> 4;
  const int l16  = lane & 15;

  v8f acc = {};
  const float* __restrict__ er = E + (size_t)(m0 + l16) * NN;
  for (int k0 = 0; k0 < NN; k0 += 4) {
    v2f a, bb;
    a[0]  = er[k0 + half * 2 + 0];
    a[1]  = er[k0 + half * 2 + 1];
    bb[0] = g[(size_t)(k0 + half * 2 + 0) * HID + n0 + l16];
    bb[1] = g[(size_t)(k0 + half * 2 + 1) * HID + n0 + l16];
    acc = wmma4(a, bb, acc);
  }
#pragma unroll
  for (int r = 0; r < 8; ++r) {
    const int m = m0 + r + half * 8;
    eph[m * HID + n0 + l16] = acc[r];
  }
}

// ---------------------------------------------------------------------------
// K3: v = rows of Ep_head normalized by their row sum (flat order)
//     grid 32, block 256
// ---------------------------------------------------------------------------
__global__ void k_vnorm(const float* __restrict__ eph, float* __restrict__ v) {
  __shared__ float red[256];
  const int row = blockIdx.x, tid = threadIdx.x;
  const float x = eph[row * HID + tid];
  red[tid] = x;
  __syncthreads();
  for (int o = 128; o > 0; o >>= 1) {
    if (tid < o) red[tid] += red[tid + o];
    __syncthreads();
  }
  v[row * HID + tid] = x / red[0];
}

// ---------------------------------------------------------------------------
// K4: u = X^T v   (128)     grid 128, block 256
// ---------------------------------------------------------------------------
__global__ void k_u(const float* __restrict__ X, const float* __restrict__ v,
                    float* __restrict__ u) {
  __shared__ float red[256];
  const int k = blockIdx.x, tid = threadIdx.x;
  float acc = 0.0f;
  for (int i = tid; i < NN; i += 256) acc = fmaf(v[i], X[(size_t)i * IN_DIM + k], acc);
  red[tid] = acc;
  __syncthreads();
  for (int o = 128; o > 0; o >>= 1) {
    if (tid < o) red[tid] += red[tid + o];
    __syncthreads();
  }
  if (tid == 0) u[k] = red[0];
}

// ---------------------------------------------------------------------------
// K5: t = W u   (256)       grid 1, block 256
// ---------------------------------------------------------------------------
__global__ void k_t(const float* __restrict__ W, const float* __restrict__ u,
                    float* __restrict__ t) {
  __shared__ float us[IN_DIM];
  const int tid = threadIdx.x;
  if (tid < IN_DIM) us[tid] = u[tid];
  __syncthreads();
  float acc = 0.0f;
#pragma unroll 8
  for (int k = 0; k < IN_DIM; ++k) acc = fmaf(W[(size_t)tid * IN_DIM + k], us[k], acc);
  t[tid] = acc;
}

// ---------------------------------------------------------------------------
// K6: out[i,:] = softmax( (v[i]/S) * t + b )     grid 8192, block 256
// ---------------------------------------------------------------------------
__global__ void k_out(const float* __restrict__ v, const float* __restrict__ t,
                      const float* __restrict__ b, float* __restrict__ out) {
  __shared__ float red[256];
  const int row = blockIdx.x, tid = threadIdx.x;
  const float c = v[row] * (1.0f / (float)NN);   // S == N (each row sums to 1)
  const float logit = fmaf(c, t[tid], b[tid]);
  red[tid] = logit;
  __syncthreads();
  for (int o = 128; o > 0; o >>= 1) {
    if (tid < o) red[tid] = fmaxf(red[tid], red[tid + o]);
    __syncthreads();
  }
  const float mx = red[0];
  __syncthreads();
  const float e = expf(logit - mx);
  red[tid] = e;
  __syncthreads();
  for (int o = 128; o > 0; o >>= 1) {
    if (tid < o) red[tid] += red[tid + o];
    __syncthreads();
  }
  out[(size_t)row * HID + tid] = e / red[0];
}

// ---------------------------------------------------------------------------
// K7: alpha = outer(v,v)/S   -- 268 MB streaming write, nontemporal float4
//     grid (8, 8192), block 256
// ---------------------------------------------------------------------------
__global__ void k_alpha(const float* __restrict__ v, float* __restrict__ alpha) {
  const int row = blockIdx.y;
  const int j4  = blockIdx.x * 256 + threadIdx.x;       // float4 column index
  const float vi = v[row] * (1.0f / (float)NN);
  const f4* __restrict__ v4 = (const f4*)v;
  f4 x = v4[j4];
  f4 o; o.x = vi * x.x; o.y = vi * x.y; o.z = vi * x.z; o.w = vi * x.w;
  __builtin_nontemporal_store(o, ((f4*)(alpha + (size_t)row * NN)) + j4);
}

// ---------------------------------------------------------------------------
extern "C" void kernel_launch(void* const* d_in, const int* in_sizes, int n_in,
                              void* d_out, int out_size, void* d_ws, size_t ws_size,
                              hipStream_t stream) {
  const float* X  = (const float*)d_in[0];   // 8192*128
  const float* E  = (const float*)d_in[1];   // 8192*8192
  const float* W  = (const float*)d_in[2];   // 256*128
  const float* b  = (const float*)d_in[3];   // 256
  const float* Ai = (const float*)d_in[4];   // 128
  const float* Aj = (const float*)d_in[5];   // 128

  float* out   = (float*)d_out;                      // 8192*256
  float* alpha = out + (size_t)NN * HID;             // 8192*8192

  float* ws   = (float*)d_ws;
  float* g    = ws;                                  // 8192*256
  float* eph  = g + (size_t)NN * HID;                // 32*256 (= 8192)
  float* v    = eph + NN;                            // 8192
  float* sbuf = v + NN;                              // 1 (padded to 8)
  float* u    = sbuf + 8;                            // 128
  float* t    = u + IN_DIM;                          // 256

  k_scalar<<<1, 256, 0, stream>>>(Ai, Aj, sbuf);
  k_gemm1 <<<dim3(NN / 16, HID / 16), 32, 0, stream>>>(X, W, b, sbuf, g);
  k_gemm2 <<<dim3(HEADROWS / 16, HID / 16), 32, 0, stream>>>(E, g, eph);
  k_vnorm <<<HEADROWS, 256, 0, stream>>>(eph, v);
  k_u     <<<IN_DIM, 256, 0, stream>>>(X, v, u);
  k_t     <<<1, 256, 0, stream>>>(W, u, t);
  k_out   <<<NN, 256, 0, stream>>>(v, t, b, out);
  k_alpha <<<dim3(NN / 4 / 256, NN), 256, 0, stream>>>(v, alpha);
}